// S4Layer_25537875542111
// MI455X (gfx1250) — compile-verified
//
#include <hip/hip_runtime.h>

// S4 layer forward for MI455X (gfx1250, wave32).
// Pipeline: [Cauchy kernel via V_WMMA_F32_16X16X4_F32] -> [2048-pt iFFT in LDS]
//           -> [4096-pt FFT of K, spectrum cached in d_ws (8 MB, L2-resident)]
//           -> [per-(b,h) 4096-pt FFT conv in LDS] -> y = conv + D*u
// Requires ws_size >= 256*4096*sizeof(float2) = 8 MB.

typedef float v2f __attribute__((ext_vector_type(2)));
typedef float v8f __attribute__((ext_vector_type(8)));

#define BATCH_SZ 8
#define H_DIM    256
#define L_SEQ    2048
#define N_DIM    64
#define NFFT     4096
#define PI_F     3.14159265358979f

// Fast hardware reciprocal (v_rcp_f32, ~1 ulp) instead of IEEE div expansion.
__device__ __forceinline__ float fast_rcp(float x) {
  return __builtin_amdgcn_rcpf(x);
}

// ---------------------------------------------------------------------------
// Kernel A: per-channel kernel construction.
//   atRoots[h,l] = c(l) * (k00 - k01*k10/(1+k11)),  k's = Cauchy sums over N
//   K = Re(ifft_2048(atRoots));  Kd = fft_4096(zeropad(K)) stored bit-reversed
// Cauchy sums are computed as a real GEMM (M=16 l-rows, K=128 re/im, N=8 out
// components) with chained v_wmma_f32_16x16x4_f32.
// ---------------------------------------------------------------------------
__global__ __launch_bounds__(256) void s4_build(
    const float* __restrict__ Lre, const float* __restrict__ Lim,
    const float* __restrict__ Pp,  const float* __restrict__ Bp,
    const float* __restrict__ Cp,  const float* __restrict__ log_step,
    float2* __restrict__ Kd)
{
  __shared__ float lamr[N_DIM], lami[N_DIM];
  __shared__ float v00r[N_DIM], v00i[N_DIM], v01r[N_DIM], v01i[N_DIM];
  __shared__ float v10r[N_DIM], v11r[N_DIM], z64[N_DIM];
  __shared__ float kmat[8][16][8];      // per-wave 16x8 WMMA result staging
  __shared__ float2 arr[L_SEQ];         // atRoots (natural order)
  __shared__ float bfr[NFFT], bfi[NFFT];// FFT working buffer

  const int h    = blockIdx.x;
  const int tid  = threadIdx.x;
  const int lane = tid & 31;
  const int wv   = tid >> 5;            // 8 waves
  const int mrow = lane & 15;           // WMMA M row / B column
  const int hi   = (lane >> 4) & 1;

  if (tid < N_DIM) {
    const int n = tid;
    lamr[n] = fminf(Lre[h * N_DIM + n], -1.0e-4f);
    lami[n] = Lim[h * N_DIM + n];
    const float Pv  = Pp[h * N_DIM + n];
    const float Bv  = Bp[h * N_DIM + n];
    const float Crv = Cp[(h * N_DIM + n) * 2 + 0];
    const float Civ = Cp[(h * N_DIM + n) * 2 + 1];
    v00r[n] =  Crv * Bv;  v00i[n] = -Civ * Bv;   // conj(C)~ * B
    v01r[n] =  Crv * Pv;  v01i[n] = -Civ * Pv;   // conj(C)~ * P
    v10r[n] =  Pv  * Bv;                          // conj(P) * B (real)
    v11r[n] =  Pv  * Pv;                          // conj(P) * P (real)
    z64[n]  = 0.0f;
  }
  __syncthreads();

  const float rstep = 2.0f * __expf(-log_step[h]);   // 2/step

  // Per-lane B-matrix column source: column mrow, K-halves lo (recip_re) /
  // hi (recip_im) of the real-embedded complex multiply.
  const float* tabLo; const float* tabHi; float sgnHi;
  switch (mrow) {
    case 0: tabLo = v00r; tabHi = v00i; sgnHi = -1.f; break; // k00_re
    case 1: tabLo = v00i; tabHi = v00r; sgnHi =  1.f; break; // k00_im
    case 2: tabLo = v01r; tabHi = v01i; sgnHi = -1.f; break; // k01_re
    case 3: tabLo = v01i; tabHi = v01r; sgnHi =  1.f; break; // k01_im
    case 4: tabLo = v10r; tabHi = z64;  sgnHi =  1.f; break; // k10_re
    case 5: tabLo = z64;  tabHi = v10r; sgnHi =  1.f; break; // k10_im
    case 6: tabLo = v11r; tabHi = z64;  sgnHi =  1.f; break; // k11_re
    case 7: tabLo = z64;  tabHi = v11r; sgnHi =  1.f; break; // k11_im
    default: tabLo = z64; tabHi = z64;  sgnHi =  1.f; break;
  }

  for (int tile = wv; tile < L_SEQ / 16; tile += 8) {
    const int l0 = tile * 16;
    // g(l) for this lane's row l = l0 + mrow (exact complex-division form,
    // matching the reference even at the near-singular l = L/2 bin).
    const float th = (2.0f * PI_F / L_SEQ) * (float)(l0 + mrow);
    float sn, cs; __sincosf(th, &sn, &cs);
    const float omr = cs, omi = -sn;                 // Omega = e^{-i th}
    const float br_ = 1.0f + omr, bi_ = omi;         // 1 + Omega
    const float ar_ = 1.0f - omr, ai_ = -omi;        // 1 - Omega
    const float invd = fast_rcp(br_ * br_ + bi_ * bi_);
    const float gr = rstep * (ar_ * br_ + ai_ * bi_) * invd;
    const float gi = rstep * (ai_ * br_ - ar_ * bi_) * invd;

    v8f acc = {};
    #pragma unroll
    for (int cc = 0; cc < 32; ++cc) {
      const int kb  = cc * 4;
      const int kkx = kb + 2 * hi;      // A/B VGPR0 K-index for this lane
      const int kky = kkx + 1;          // A/B VGPR1 K-index
      // A fragment: recip_re (kk<64) or recip_im (kk>=64) of 1/(g - lambda_n)
      const int nx = kkx & 63;
      const float drx = gr - lamr[nx], dix = gi - lami[nx];
      const float rrx = fast_rcp(drx * drx + dix * dix);
      const float ax  = (kkx < 64) ? (drx * rrx) : (-dix * rrx);
      const int ny = kky & 63;
      const float dry = gr - lamr[ny], diy = gi - lami[ny];
      const float rry = fast_rcp(dry * dry + diy * diy);
      const float ay  = (kky < 64) ? (dry * rry) : (-diy * rry);
      // B fragment
      const float bx = (kkx < 64) ? tabLo[nx] : sgnHi * tabHi[nx];
      const float by = (kky < 64) ? tabLo[ny] : sgnHi * tabHi[ny];
      v2f av; av.x = ax; av.y = ay;
      v2f bv; bv.x = bx; bv.y = by;
      acc = __builtin_amdgcn_wmma_f32_16x16x4_f32(
          /*neg_a=*/false, av, /*neg_b=*/false, bv,
          /*c_mod=*/(short)0, acc, /*reuse_a=*/false, /*reuse_b=*/false);
    }

    __syncthreads();
    if (mrow < 8) {
      #pragma unroll
      for (int r = 0; r < 8; ++r)
        kmat[wv][r + 8 * hi][mrow] = acc[r];   // D layout: M = r + 8*hi
    }
    __syncthreads();

    if (lane < 16) {
      const int l2 = l0 + lane;
      const float k00r_ = kmat[wv][lane][0], k00i_ = kmat[wv][lane][1];
      const float k01r_ = kmat[wv][lane][2], k01i_ = kmat[wv][lane][3];
      const float k10r_ = kmat[wv][lane][4], k10i_ = kmat[wv][lane][5];
      const float k11r_ = kmat[wv][lane][6], k11i_ = kmat[wv][lane][7];
      const float th2 = (2.0f * PI_F / L_SEQ) * (float)l2;
      float sn2, cs2; __sincosf(th2, &sn2, &cs2);
      const float b2r = 1.0f + cs2, b2i = -sn2;        // 1 + Omega
      const float inv2 = fast_rcp(b2r * b2r + b2i * b2i);
      const float cwr =  2.0f * b2r * inv2;            // c = 2/(1+Omega)
      const float cwi = -2.0f * b2i * inv2;
      const float w1r = 1.0f + k11r_, w1i = k11i_;
      const float invw = fast_rcp(w1r * w1r + w1i * w1i);
      const float numr = k01r_ * k10r_ - k01i_ * k10i_;
      const float numi = k01r_ * k10i_ + k01i_ * k10r_;
      const float qr = (numr * w1r + numi * w1i) * invw;
      const float qi = (numi * w1r - numr * w1i) * invw;
      const float sr  = k00r_ - qr, si2 = k00i_ - qi;
      arr[l2] = make_float2(cwr * sr - cwi * si2, cwr * si2 + cwi * sr);
    }
  }
  __syncthreads();

  // ---- 2048-pt inverse FFT (DIT, bit-reversed input load) -> K (natural) ----
  for (int i = tid; i < L_SEQ; i += 256) {
    const unsigned j = __brev((unsigned)i) >> 21;    // 11-bit reverse
    const float2 v = arr[j];
    bfr[i] = v.x; bfi[i] = v.y;
  }
  __syncthreads();
  for (int p = 0; p <= 10; ++p) {
    const int m = 1 << p;
    const float angs = PI_F / (float)m;              // +angle: inverse
    for (int j = tid; j < L_SEQ / 2; j += 256) {
      const int t = j & (m - 1);
      const int base = ((j >> p) << (p + 1)) + t;
      const int pr = base + m;
      float wsn, wc; __sincosf(angs * (float)t, &wsn, &wc);
      const float xr = bfr[pr], xi = bfi[pr];
      const float tr  = xr * wc - xi * wsn;
      const float ti2 = xr * wsn + xi * wc;
      const float ur = bfr[base], ui = bfi[base];
      bfr[base] = ur + tr; bfi[base] = ui + ti2;
      bfr[pr]   = ur - tr; bfi[pr]   = ui - ti2;
    }
    __syncthreads();
  }

  // K real part (scaled), zero-pad to 4096
  float kvv[8];
  #pragma unroll
  for (int q = 0; q < 8; ++q) kvv[q] = bfr[tid + 256 * q] * (1.0f / 2048.0f);
  __syncthreads();
  #pragma unroll
  for (int q = 0; q < 8; ++q) {
    const int t = tid + 256 * q;
    bfr[t] = kvv[q];        bfi[t] = 0.0f;
    bfr[t + 2048] = 0.0f;   bfi[t + 2048] = 0.0f;
  }
  __syncthreads();

  // ---- 4096-pt forward FFT (DIF, natural in -> bit-reversed out) ----
  for (int p = 11; p >= 0; --p) {
    const int m = 1 << p;
    const float angs = -PI_F / (float)m;
    for (int j = tid; j < NFFT / 2; j += 256) {
      const int t = j & (m - 1);
      const int base = ((j >> p) << (p + 1)) + t;
      const int pr = base + m;
      float wsn, wc; __sincosf(angs * (float)t, &wsn, &wc);
      const float ur = bfr[base], ui = bfi[base];
      const float xr = bfr[pr],  xi = bfi[pr];
      const float dr = ur - xr,  di = ui - xi;
      bfr[base] = ur + xr;       bfi[base] = ui + xi;
      bfr[pr] = dr * wc - di * wsn;
      bfi[pr] = dr * wsn + di * wc;
    }
    __syncthreads();
  }

  float2* out = Kd + (size_t)h * NFFT;
  for (int i = tid; i < NFFT; i += 256)
    out[i] = make_float2(bfr[i], bfi[i]);
}

// ---------------------------------------------------------------------------
// Kernel B: per-(b,h) FFT convolution. Both spectra are in identical
// bit-reversed order (DIF fwd) so the pointwise product needs no reorder;
// DIT inverse consumes bit-reversed and emits natural order.
// ---------------------------------------------------------------------------
__global__ __launch_bounds__(256) void s4_conv(
    const float* __restrict__ u, const float* __restrict__ Dp,
    const float2* __restrict__ Kd, float* __restrict__ y)
{
  __shared__ float bfr[NFFT], bfi[NFFT];
  __shared__ float twr[NFFT / 2], twi[NFFT / 2];   // e^{-2pi i k/4096}
  const int h = blockIdx.x, b = blockIdx.y;
  const int tid = threadIdx.x;
  const size_t ubase = ((size_t)b * H_DIM + h) * L_SEQ;

  for (int k = tid; k < NFFT / 2; k += 256) {
    float s, c; __sincosf(-(2.0f * PI_F / NFFT) * (float)k, &s, &c);
    twr[k] = c; twi[k] = s;
  }
  float uv[8];
  #pragma unroll
  for (int q = 0; q < 8; ++q) {
    const int t = tid + 256 * q;
    const float v = u[ubase + t];
    uv[q] = v;
    bfr[t] = v;            bfi[t] = 0.0f;
    bfr[t + 2048] = 0.0f;  bfi[t + 2048] = 0.0f;
  }
  __syncthreads();

  // forward DIF 4096 (table twiddles: idx = t << (11 - p))
  for (int p = 11; p >= 0; --p) {
    const int m = 1 << p; const int sh = 11 - p;
    for (int j = tid; j < NFFT / 2; j += 256) {
      const int t = j & (m - 1);
      const int base = ((j >> p) << (p + 1)) + t;
      const int pr = base + m;
      const int ti = t << sh;
      const float wc = twr[ti], wsn = twi[ti];
      const float ur = bfr[base], ui = bfi[base];
      const float xr = bfr[pr],  xi = bfi[pr];
      const float dr = ur - xr,  di = ui - xi;
      bfr[base] = ur + xr;       bfi[base] = ui + xi;
      bfr[pr] = dr * wc - di * wsn;
      bfi[pr] = dr * wsn + di * wc;
    }
    __syncthreads();
  }

  // pointwise product with kernel spectrum (+ inverse-FFT scale)
  const float2* kd = Kd + (size_t)h * NFFT;
  const float scale = 1.0f / (float)NFFT;
  for (int i = tid; i < NFFT; i += 256) {
    const float2 kk = kd[i];
    const float xr = bfr[i], xi = bfi[i];
    bfr[i] = (xr * kk.x - xi * kk.y) * scale;
    bfi[i] = (xr * kk.y + xi * kk.x) * scale;
  }
  __syncthreads();

  // inverse DIT 4096 (conjugate table)
  for (int p = 0; p <= 11; ++p) {
    const int m = 1 << p; const int sh = 11 - p;
    for (int j = tid; j < NFFT / 2; j += 256) {
      const int t = j & (m - 1);
      const int base = ((j >> p) << (p + 1)) + t;
      const int pr = base + m;
      const int ti = t << sh;
      const float wc = twr[ti], wsn = -twi[ti];
      const float xr = bfr[pr], xi = bfi[pr];
      const float tr  = xr * wc - xi * wsn;
      const float ti2 = xr * wsn + xi * wc;
      const float ur = bfr[base], ui = bfi[base];
      bfr[base] = ur + tr; bfi[base] = ui + ti2;
      bfr[pr]   = ur - tr; bfi[pr]   = ui - ti2;
    }
    __syncthreads();
  }

  const float Dv = Dp[h];
  #pragma unroll
  for (int q = 0; q < 8; ++q) {
    const int t = tid + 256 * q;
    y[ubase + t] = bfr[t] + Dv * uv[q];
  }
}

extern "C" void kernel_launch(void* const* d_in, const int* in_sizes, int n_in,
                              void* d_out, int out_size, void* d_ws, size_t ws_size,
                              hipStream_t stream) {
  const float* u   = (const float*)d_in[0];
  const float* Lre = (const float*)d_in[1];
  const float* Lim = (const float*)d_in[2];
  const float* P   = (const float*)d_in[3];
  const float* B   = (const float*)d_in[4];
  const float* C   = (const float*)d_in[5];
  const float* D   = (const float*)d_in[6];
  const float* ls  = (const float*)d_in[7];
  float* y = (float*)d_out;
  float2* Kd = (float2*)d_ws;   // 256 * 4096 * 8B = 8 MB of scratch

  s4_build<<<dim3(H_DIM), dim3(256), 0, stream>>>(Lre, Lim, P, B, C, ls, Kd);
  s4_conv<<<dim3(H_DIM, BATCH_SZ), dim3(256), 0, stream>>>(u, D, Kd, y);
}